// GraphGATNet_11175504904833
// MI455X (gfx1250) — compile-verified
//
#include <hip/hip_runtime.h>
#include <hip/hip_bf16.h>

// ---------------------------------------------------------------------------
// GraphGATNet on MI455X (gfx1250, wave32)
//   GEMMs: fp32-exact V_WMMA_F32_16X16X4_F32, 16x64 output strip per wave
//          (4 accumulators reuse one A fragment -> 4x less A traffic)
//   Edge phase: CSR build once, per-dst-node two-pass segment softmax
//          with coalesced gathers and fused bias (+ReLU) epilogues.
// ---------------------------------------------------------------------------

typedef float v2f __attribute__((ext_vector_type(2)));
typedef float v8f __attribute__((ext_vector_type(8)));

#define NEG_SLOPE 0.2f

#define WMMA_F32(a, b, c) \
    __builtin_amdgcn_wmma_f32_16x16x4_f32(false, (a), false, (b), (short)0, (c), false, false)

// ---------------- GEMM: C[M,Nout] = A[M,K] @ B[K,Nout], fp32 WMMA ----------
// One 16(M) x 64(N) strip per wave: 4 WMMAs per k-step share the A fragment.
// Requires: M % 16 == 0, Nout % 64 == 0, K % 4 == 0.
__global__ __launch_bounds__(256) void gemm_wmma_f32(
        const float* __restrict__ A, const float* __restrict__ B,
        float* __restrict__ C, int M, int K, int Nout) {
    const int lane = threadIdx.x & 31;
    const int wave = threadIdx.x >> 5;
    const int tilesN = Nout >> 6;                 // 64-wide strips
    const int tiles  = (M >> 4) * tilesN;
    const int tile   = blockIdx.x * 8 + wave;     // 8 waves / block
    if (tile >= tiles) return;                    // wave-uniform branch

    const int tm = tile / tilesN;
    const int tn = tile - tm * tilesN;
    const int m0 = tm << 4;
    const int n0 = tn << 6;

    const int l15 = lane & 15;
    const int hi  = lane >> 4;                    // 0: K={0,1}, 1: K={2,3}

    // ISA 7.12.2 layouts for 32-bit A (16x4) and B (4x16)
    const float* arow  = A + (size_t)(m0 + l15) * K + 2 * hi;
    const float* bbase = B + (size_t)(2 * hi) * Nout + n0 + l15;

    v8f c0 = {}, c1 = {}, c2 = {}, c3 = {};
    for (int k = 0; k < K; k += 4) {
        v2f a;
        a.x = arow[k];
        a.y = arow[k + 1];
        const float* bk = bbase + (size_t)k * Nout;
        v2f b0, b1, b2, b3;
        b0.x = bk[0];   b0.y = bk[(size_t)Nout + 0];
        b1.x = bk[16];  b1.y = bk[(size_t)Nout + 16];
        b2.x = bk[32];  b2.y = bk[(size_t)Nout + 32];
        b3.x = bk[48];  b3.y = bk[(size_t)Nout + 48];
        c0 = WMMA_F32(a, b0, c0);                 // independent D regs:
        c1 = WMMA_F32(a, b1, c1);                 // pipeline back-to-back,
        c2 = WMMA_F32(a, b2, c2);                 // no RAW hazard NOPs
        c3 = WMMA_F32(a, b3, c3);
    }

    // C layout: VGPR r -> row m0 + r + 8*hi, col = strip col + l15
    float* crow = C + (size_t)(m0 + 8 * hi) * Nout + n0 + l15;
#pragma unroll
    for (int r = 0; r < 8; ++r) {
        crow[(size_t)r * Nout +  0] = c0[r];
        crow[(size_t)r * Nout + 16] = c1[r];
        crow[(size_t)r * Nout + 32] = c2[r];
        crow[(size_t)r * Nout + 48] = c3[r];
    }
}

// ---------------- attention scores: a = sum_c h[n,h,c]*att[h,c] ------------
// one wave32 per (node, head); C fixed at 128 (both layers).
__global__ __launch_bounds__(256) void attn_scores(
        const float* __restrict__ hmat,
        const float* __restrict__ att_s, const float* __restrict__ att_d,
        float* __restrict__ a_src, float* __restrict__ a_dst, int n, int H) {
    const int C = 128;
    const int lane = threadIdx.x & 31;
    const int wave = threadIdx.x >> 5;
    const int pair = blockIdx.x * 8 + wave;       // 256 threads = 8 waves
    if (pair >= n * H) return;                    // wave-uniform
    const int h = pair % H;

    const float4* hv = (const float4*)(hmat + (size_t)pair * C);
    const float4* as = (const float4*)(att_s + (size_t)h * C);
    const float4* ad = (const float4*)(att_d + (size_t)h * C);
    float4 x  = hv[lane];
    float4 s4 = as[lane];
    float4 d4 = ad[lane];
    float ss = x.x * s4.x + x.y * s4.y + x.z * s4.z + x.w * s4.w;
    float dd = x.x * d4.x + x.y * d4.y + x.z * d4.z + x.w * d4.w;
#pragma unroll
    for (int off = 16; off; off >>= 1) {
        ss += __shfl_xor(ss, off, 32);
        dd += __shfl_xor(dd, off, 32);
    }
    if (lane == 0) { a_src[pair] = ss; a_dst[pair] = dd; }
}

// ---------------- CSR build ------------------------------------------------
// edges: e < E -> (src=ei[e], dst=ei[E+e]); e >= E -> self loop (e-E, e-E)
__global__ __launch_bounds__(256) void count_deg(
        const int* __restrict__ ei, int* __restrict__ deg, int E, int n) {
    const int e = blockIdx.x * blockDim.x + threadIdx.x;
    if (e >= E + n) return;
    const int d = (e < E) ? ei[E + e] : (e - E);
    atomicAdd(&deg[d], 1);
}

__global__ __launch_bounds__(1024) void scan_rowstart(
        const int* __restrict__ deg, int* __restrict__ row_start, int n) {
    __shared__ int partial[1024];
    const int t = threadIdx.x;
    const int per = (n + 1023) / 1024;
    const int begin = t * per;
    const int end = (begin + per < n) ? (begin + per) : n;
    int s = 0;
    for (int i = begin; i < end; ++i) s += deg[i];
    partial[t] = s;
    __syncthreads();
    for (int off = 1; off < 1024; off <<= 1) {
        int v = (t >= off) ? partial[t - off] : 0;
        __syncthreads();
        partial[t] += v;
        __syncthreads();
    }
    int running = (t == 0) ? 0 : partial[t - 1];
    for (int i = begin; i < end; ++i) { row_start[i] = running; running += deg[i]; }
    if (t == 1023) row_start[n] = partial[1023];
}

__global__ __launch_bounds__(256) void scatter_csr(
        const int* __restrict__ ei, const int* __restrict__ row_start,
        int* __restrict__ cursor, int* __restrict__ csr_src, int E, int n) {
    const int e = blockIdx.x * blockDim.x + threadIdx.x;
    if (e >= E + n) return;
    int s, d;
    if (e < E) { s = ei[e]; d = ei[E + e]; } else { s = e - E; d = s; }
    const int slot = row_start[d] + atomicAdd(&cursor[d], 1);
    csr_src[slot] = s;
}

// ---------------- segment softmax + aggregation ----------------------------
// blockDim = H*128, one block per destination node. Fused bias (+ReLU).
template <int H, bool RELU>
__global__ __launch_bounds__(H * 128) void aggregate(
        const float* __restrict__ hprev,
        const float* __restrict__ a_src, const float* __restrict__ a_dst,
        const int* __restrict__ row_start, const int* __restrict__ csr_src,
        const float* __restrict__ bias, float* __restrict__ out, int n) {
    constexpr int C = 128;
    const int i = blockIdx.x;
    const int t = threadIdx.x;            // 0 .. H*C-1
    const int h = t / C;
    const int c = t - h * C;
    const float ad = a_dst[(size_t)i * H + h];
    const int row = row_start[i];
    const int end = row_start[i + 1];
    const float* hb = hprev + (size_t)h * C + c;   // gather base for this lane

    // pass 1: per-head max logit (a_src loads broadcast across the head)
    float m = -3.4e38f;
    for (int j = row; j < end; ++j) {
        const int s = csr_src[j];
        float v = a_src[(size_t)s * H + h] + ad;
        v = (v > 0.f) ? v : NEG_SLOPE * v;
        m = fmaxf(m, v);
    }
    // pass 2: weights + weighted feature sum (coalesced hprev reads)
    float denom = 0.f, acc = 0.f;
    for (int j = row; j < end; ++j) {
        const int s = csr_src[j];
        float v = a_src[(size_t)s * H + h] + ad;
        v = (v > 0.f) ? v : NEG_SLOPE * v;
        const float w = __expf(v - m);
        denom += w;
        acc += w * hb[(size_t)s * (H * C)];
    }
    float val = acc / (denom + 1e-16f) + bias[h * C + c];
    if (RELU) val = fmaxf(val, 0.f);
    out[(size_t)i * (H * C) + t] = val;
}

// ---------------------------------------------------------------------------
extern "C" void kernel_launch(void* const* d_in, const int* in_sizes, int n_in,
                              void* d_out, int out_size, void* d_ws, size_t ws_size,
                              hipStream_t stream) {
    (void)n_in; (void)out_size; (void)ws_size;
    const float* x        = (const float*)d_in[0];
    const int*   ei       = (const int*)d_in[1];
    const float* W1       = (const float*)d_in[2];
    const float* att_src1 = (const float*)d_in[3];
    const float* att_dst1 = (const float*)d_in[4];
    const float* b1       = (const float*)d_in[5];
    const float* W2       = (const float*)d_in[6];
    const float* att_src2 = (const float*)d_in[7];
    const float* att_dst2 = (const float*)d_in[8];
    const float* b2       = (const float*)d_in[9];

    const int IN = 384, H = 8, C = 128, HC = H * C, OUT = 128;
    const int N = in_sizes[0] / IN;
    const int E = in_sizes[1] / 2;
    const int Etot = E + N;

    // workspace carve-out (256B aligned slices)
    char* ws = (char*)d_ws;
    auto carve = [&](size_t bytes) -> void* {
        void* p = (void*)ws;
        ws += (bytes + 255) & ~(size_t)255;
        return p;
    };
    float* h1        = (float*)carve((size_t)N * HC * 4);  // reused as h2
    float* x2        = (float*)carve((size_t)N * HC * 4);
    float* as1       = (float*)carve((size_t)N * H * 4);   // reused for layer2
    float* ad1       = (float*)carve((size_t)N * H * 4);   // reused for layer2
    int*   deg       = (int*)carve((size_t)N * 4);         // reused as cursor
    int*   row_start = (int*)carve((size_t)(N + 1) * 4);
    int*   csr_src   = (int*)carve((size_t)Etot * 4);
    float* out       = (float*)d_out;

    // ---- CSR build (shared by both layers) ----
    hipMemsetAsync(deg, 0, (size_t)N * 4, stream);
    count_deg<<<(Etot + 255) / 256, 256, 0, stream>>>(ei, deg, E, N);
    scan_rowstart<<<1, 1024, 0, stream>>>(deg, row_start, N);
    hipMemsetAsync(deg, 0, (size_t)N * 4, stream);  // reuse as cursor
    scatter_csr<<<(Etot + 255) / 256, 256, 0, stream>>>(ei, row_start, deg, csr_src, E, N);

    // ---- layer 1 ----
    {
        const int tiles = (N >> 4) * (HC >> 6);   // 16x64 strips
        gemm_wmma_f32<<<(tiles + 7) / 8, 256, 0, stream>>>(x, W1, h1, N, IN, HC);
        attn_scores<<<(N * H + 7) / 8, 256, 0, stream>>>(h1, att_src1, att_dst1, as1, ad1, N, H);
        aggregate<8, true><<<N, H * C, 0, stream>>>(h1, as1, ad1, row_start, csr_src, b1, x2, N);
    }

    // ---- layer 2 (h2 aliases h1 buffer; scores alias as1/ad1) ----
    {
        float* h2 = h1;
        const int tiles = (N >> 4) * (OUT >> 6);
        gemm_wmma_f32<<<(tiles + 7) / 8, 256, 0, stream>>>(x2, W2, h2, N, HC, OUT);
        attn_scores<<<(N * 1 + 7) / 8, 256, 0, stream>>>(h2, att_src2, att_dst2, as1, ad1, N, 1);
        aggregate<1, false><<<N, 1 * C, 0, stream>>>(h2, as1, ad1, row_start, csr_src, b2, out, N);
    }
}